// SinkhornDistance_15599321219680
// MI455X (gfx1250) — compile-verified
//
#include <hip/hip_runtime.h>
#include <math.h>

#define P        2048
#define NB       8
#define TILES    (P / 16)            // 128 tiles per dimension
#define SPLIT    8                   // column-segment split per 16-row stripe
#define SEGTILES (TILES / SPLIT)     // 16 tiles per wave sweep
#define WAVES    (NB * TILES * SPLIT)// 8192 waves per sweep pass
#define EPS_F    1e-4f
#define INV_EPS  1e4f
#define LOG_MARG (-7.6246189861593985f)   // -log(2048)

typedef __attribute__((ext_vector_type(2))) float v2f;
typedef __attribute__((ext_vector_type(8))) float v8f;

// ---------------------------------------------------------------------------
// Prep: build WMMA operand packs in lane layout, squared norms, zero duals.
// Pack layout: pack[(n*TILES + t)*32 + lane] = {pts[k0], pts[k0+1]} where
// k0 = 2*(lane>>4), K=3 padded with 0; row/col index = t*16 + (lane&15).
// ---------------------------------------------------------------------------
__global__ void sinkhorn_prep(const float* __restrict__ x, const float* __restrict__ y,
                              v2f* __restrict__ xpack, v2f* __restrict__ ypack,
                              float* __restrict__ x2, float* __restrict__ y2,
                              float* __restrict__ u, float* __restrict__ v) {
    int idx = blockIdx.x * blockDim.x + threadIdx.x;   // exactly NB*P threads
    int n  = idx / P;
    int j  = idx % P;
    int t  = j >> 4;
    int lm = j & 15;

    const float* xp = x + (size_t)idx * 3;
    const float* yp = y + (size_t)idx * 3;
    float x0 = xp[0], x1 = xp[1], x2v = xp[2];
    float y0 = yp[0], y1 = yp[1], y2v = yp[2];

    x2[idx] = x0 * x0 + x1 * x1 + x2v * x2v;
    y2[idx] = y0 * y0 + y1 * y1 + y2v * y2v;
    u[idx] = 0.0f;
    v[idx] = 0.0f;

    size_t pb = ((size_t)n * TILES + t) * 32 + lm;
    v2f lo, hi;
    lo.x = x0; lo.y = x1;  hi.x = x2v; hi.y = 0.0f;
    xpack[pb] = lo;  xpack[pb + 16] = hi;
    lo.x = y0; lo.y = y1;  hi.x = y2v; hi.y = 0.0f;
    ypack[pb] = lo;  ypack[pb + 16] = hi;
}

// ---------------------------------------------------------------------------
// Split half-iteration: wave owns (batch n, 16-row stripe rt, segment seg of
// 16 column tiles). Streams 16x16x4 f32 WMMA Gram tiles fused with online
// logsumexp; writes per-segment (m, s) partials.
// ---------------------------------------------------------------------------
__global__ __launch_bounds__(256) void sinkhorn_pass(
        const v2f* __restrict__ packA, const float* __restrict__ nA,
        const v2f* __restrict__ packB, const float* __restrict__ nB,
        const float* __restrict__ dualB,
        float* __restrict__ mpart, float* __restrict__ spart) {
    const int lane = threadIdx.x & 31;
    const int wave = blockIdx.x * 8 + (threadIdx.x >> 5);   // 0..WAVES-1
    const int n    = wave / (TILES * SPLIT);
    const int rem  = wave % (TILES * SPLIT);
    const int rt   = rem / SPLIT;                           // row tile
    const int seg  = rem % SPLIT;                           // column segment
    const int half = lane >> 4;
    const int lm   = lane & 15;
    const size_t base = (size_t)n * P;

    const v2f a = packA[((size_t)n * TILES + rt) * 32 + lane];

    float xs[8];
#pragma unroll
    for (int c = 0; c < 8; ++c) xs[c] = nA[base + rt * 16 + c + 8 * half];

    float m[8], s[8];
#pragma unroll
    for (int c = 0; c < 8; ++c) { m[c] = -INFINITY; s[c] = 0.0f; }

    const v2f*   bp = packB + ((size_t)n * TILES + seg * SEGTILES) * 32 + lane;
    const float* nb = nB    + base + seg * SEGTILES * 16 + lm;
    const float* db = dualB + base + seg * SEGTILES * 16 + lm;

    for (int t = 0; t < SEGTILES; ++t) {
        v2f   b   = bp[t * 32];
        float y2j = nb[t * 16];
        float dvj = db[t * 16];

        v8f c0 = {};
        v8f d = __builtin_amdgcn_wmma_f32_16x16x4_f32(
            false, a, false, b, (short)0, c0, false, false);

#pragma unroll
        for (int c = 0; c < 8; ++c) {
            float d2 = fmaxf(xs[c] + y2j - 2.0f * d[c], 0.0f);
            float M  = __builtin_amdgcn_sqrtf(d2);
            float z  = (dvj - M) * INV_EPS;
            float mn = fmaxf(m[c], z);
            s[c] = s[c] * __expf(m[c] - mn) + __expf(z - mn);
            m[c] = mn;
        }
    }

    // Combine across the 16 lanes of each half (masks < 16 stay in-half).
#pragma unroll
    for (int c = 0; c < 8; ++c) {
#pragma unroll
        for (int off = 1; off < 16; off <<= 1) {
            float mo = __shfl_xor(m[c], off, 32);
            float so = __shfl_xor(s[c], off, 32);
            float mn = fmaxf(m[c], mo);
            s[c] = s[c] * __expf(m[c] - mn) + so * __expf(mo - mn);
            m[c] = mn;
        }
    }

    if (lm == 0) {
#pragma unroll
        for (int c = 0; c < 8; ++c) {
            int row = rt * 16 + c + 8 * half;
            size_t o = (base + row) * SPLIT + seg;
            mpart[o] = m[c];
            spart[o] = s[c];
        }
    }
}

// ---------------------------------------------------------------------------
// Merge the SPLIT per-segment (m,s) partials of each row -> dual potential.
// ---------------------------------------------------------------------------
__global__ void sinkhorn_combine(const float* __restrict__ mpart,
                                 const float* __restrict__ spart,
                                 float* __restrict__ out) {
    int i = blockIdx.x * blockDim.x + threadIdx.x;   // 0..NB*P-1
    const float* mp = mpart + (size_t)i * SPLIT;
    const float* sp = spart + (size_t)i * SPLIT;
    float mm = -INFINITY;
#pragma unroll
    for (int k = 0; k < SPLIT; ++k) mm = fmaxf(mm, mp[k]);
    float ss = 0.0f;
#pragma unroll
    for (int k = 0; k < SPLIT; ++k) ss += sp[k] * __expf(mp[k] - mm);
    out[i] = EPS_F * (LOG_MARG - (mm + __logf(ss)));
}

// ---------------------------------------------------------------------------
// Final cost partials, same split: sum exp((u_i + v_j - M)/eps) * M.
// ---------------------------------------------------------------------------
__global__ __launch_bounds__(256) void sinkhorn_final(
        const v2f* __restrict__ packA, const float* __restrict__ nA,
        const v2f* __restrict__ packB, const float* __restrict__ nB,
        const float* __restrict__ u, const float* __restrict__ v,
        float* __restrict__ partials) {
    const int lane = threadIdx.x & 31;
    const int wave = blockIdx.x * 8 + (threadIdx.x >> 5);
    const int n    = wave / (TILES * SPLIT);
    const int rem  = wave % (TILES * SPLIT);
    const int rt   = rem / SPLIT;
    const int seg  = rem % SPLIT;
    const int half = lane >> 4;
    const int lm   = lane & 15;
    const size_t base = (size_t)n * P;

    const v2f a = packA[((size_t)n * TILES + rt) * 32 + lane];

    float xs[8], us[8];
#pragma unroll
    for (int c = 0; c < 8; ++c) {
        int row = rt * 16 + c + 8 * half;
        xs[c] = nA[base + row];
        us[c] = u[base + row];
    }

    const v2f*   bp = packB + ((size_t)n * TILES + seg * SEGTILES) * 32 + lane;
    const float* nb = nB + base + seg * SEGTILES * 16 + lm;
    const float* db = v  + base + seg * SEGTILES * 16 + lm;

    float acc = 0.0f;
    for (int t = 0; t < SEGTILES; ++t) {
        v2f   b   = bp[t * 32];
        float y2j = nb[t * 16];
        float dvj = db[t * 16];

        v8f c0 = {};
        v8f d = __builtin_amdgcn_wmma_f32_16x16x4_f32(
            false, a, false, b, (short)0, c0, false, false);

#pragma unroll
        for (int c = 0; c < 8; ++c) {
            float d2 = fmaxf(xs[c] + y2j - 2.0f * d[c], 0.0f);
            float M  = __builtin_amdgcn_sqrtf(d2);
            float p  = __expf((us[c] + dvj - M) * INV_EPS);
            acc = fmaf(p, M, acc);
        }
    }

#pragma unroll
    for (int off = 1; off < 32; off <<= 1)
        acc += __shfl_xor(acc, off, 32);

    if (lane == 0) partials[wave] = acc;
}

// ---------------------------------------------------------------------------
// Deterministic fixed-order sum of TILES*SPLIT partials per batch.
// ---------------------------------------------------------------------------
__global__ void sinkhorn_reduce(const float* __restrict__ partials, float* __restrict__ out) {
    __shared__ float sm[256];
    const int t = threadIdx.x;
    const int n = blockIdx.x;
    const float* pp = partials + (size_t)n * (TILES * SPLIT);
    float a = pp[t] + pp[t + 256];
    float b = pp[t + 512] + pp[t + 768];
    sm[t] = a + b;
    __syncthreads();
    for (int off = 128; off > 0; off >>= 1) {
        if (t < off) sm[t] += sm[t + off];
        __syncthreads();
    }
    if (t == 0) out[n] = sm[0];
}

// ---------------------------------------------------------------------------
extern "C" void kernel_launch(void* const* d_in, const int* in_sizes, int n_in,
                              void* d_out, int out_size, void* d_ws, size_t ws_size,
                              hipStream_t stream) {
    const float* x = (const float*)d_in[0];   // [8,2048,3] f32
    const float* y = (const float*)d_in[1];   // [8,2048,3] f32
    float* out = (float*)d_out;               // [8] f32

    // Workspace layout (floats), ~1.8 MB total:
    float* ws = (float*)d_ws;
    size_t off = 0;
    v2f*   xpack = (v2f*)(ws + off);  off += (size_t)2 * NB * TILES * 32;  // 65536
    v2f*   ypack = (v2f*)(ws + off);  off += (size_t)2 * NB * TILES * 32;  // 65536
    float* x2    = ws + off;          off += (size_t)NB * P;               // 16384
    float* y2    = ws + off;          off += (size_t)NB * P;
    float* u     = ws + off;          off += (size_t)NB * P;
    float* v     = ws + off;          off += (size_t)NB * P;
    float* mpart = ws + off;          off += (size_t)NB * P * SPLIT;       // 131072
    float* spart = ws + off;          off += (size_t)NB * P * SPLIT;
    float* fpart = ws + off;          off += (size_t)WAVES;                // 8192

    sinkhorn_prep<<<(NB * P) / 256, 256, 0, stream>>>(x, y, xpack, ypack, x2, y2, u, v);

    // 3 Sinkhorn iterations: u from v, then v from u (x/y roles swapped).
    for (int it = 0; it < 3; ++it) {
        sinkhorn_pass<<<WAVES / 8, 256, 0, stream>>>(xpack, x2, ypack, y2, v, mpart, spart);
        sinkhorn_combine<<<(NB * P) / 256, 256, 0, stream>>>(mpart, spart, u);
        sinkhorn_pass<<<WAVES / 8, 256, 0, stream>>>(ypack, y2, xpack, x2, u, mpart, spart);
        sinkhorn_combine<<<(NB * P) / 256, 256, 0, stream>>>(mpart, spart, v);
    }

    sinkhorn_final<<<WAVES / 8, 256, 0, stream>>>(xpack, x2, ypack, y2, u, v, fpart);
    sinkhorn_reduce<<<NB, 256, 0, stream>>>(fpart, out);
}